// EfficientMemoryGEMM_22720376996351
// MI455X (gfx1250) — compile-verified
//
#include <hip/hip_runtime.h>
#include <hip/hip_bf16.h>
#include <math.h>

typedef __attribute__((ext_vector_type(2))) float v2f;
typedef __attribute__((ext_vector_type(8))) float v8f;

#define TOT_ELEMS 4194304u      // 32*2048*64 elements per input tensor
#define MAT_ELEMS 131072u       // 2048*64
#define GEMM_ELEMS 134217728ull // 32*2048*2048
#define QMAXF 127.0f
#define QTARGET 41943u          // ~1% of 4194304 above threshold

// ======================= GEMM: C[b] = A[b](2048x64) @ B[b](64x2048) =======================
// grid (32, 16, 32) = (N/64, M/128, batch); block 256 = 8 waves; wave -> 16x64 C strip.
__global__ __launch_bounds__(256) void gemm_f32_wmma(const float* __restrict__ A,
                                                     const float* __restrict__ B,
                                                     float* __restrict__ C) {
  const int lane = threadIdx.x & 31;
  const int wave = threadIdx.x >> 5;
  const int ll = lane & 15;      // lane low: M (for A) or N (for B/C)
  const int lh = lane >> 4;      // lane half selects K pair (A/B) or M+8 (C)
  const int b  = blockIdx.z;
  const int m0 = blockIdx.y * 128 + wave * 16;
  const int n0 = blockIdx.x * 64;

  const float* Ab = A + (size_t)b * (2048 * 64);
  const float* Bb = B + (size_t)b * (64 * 2048);
  float*       Cb = C + (size_t)b * (2048ull * 2048ull);

  // A operand: lanes 0-15 hold K0(v0),K1(v1); lanes 16-31 hold K2(v0),K3(v1)
  const float* arow  = Ab + (size_t)(m0 + ll) * 64 + 2 * lh;
  // B operand mirrored on N
  const float* bbase = Bb + (size_t)(2 * lh) * 2048 + n0 + ll;

  v8f acc0 = {}, acc1 = {}, acc2 = {}, acc3 = {};
#pragma unroll
  for (int kk = 0; kk < 16; ++kk) {                 // K = 64 in steps of 4
    v2f a;
    a.x = arow[kk * 4];
    a.y = arow[kk * 4 + 1];
    const float* bp = bbase + (size_t)(kk * 4) * 2048;
    v2f b0, b1, b2, b3;
    b0.x = bp[0];  b0.y = bp[2048];
    b1.x = bp[16]; b1.y = bp[2048 + 16];
    b2.x = bp[32]; b2.y = bp[2048 + 32];
    b3.x = bp[48]; b3.y = bp[2048 + 48];
    acc0 = __builtin_amdgcn_wmma_f32_16x16x4_f32(false, a, false, b0, (short)0, acc0, false, false);
    acc1 = __builtin_amdgcn_wmma_f32_16x16x4_f32(false, a, false, b1, (short)0, acc1, false, false);
    acc2 = __builtin_amdgcn_wmma_f32_16x16x4_f32(false, a, false, b2, (short)0, acc2, false, false);
    acc3 = __builtin_amdgcn_wmma_f32_16x16x4_f32(false, a, false, b3, (short)0, acc3, false, false);
  }
  // C/D layout: VGPR r -> row (m0 + r + 8*lh), col n0+16j+ll
  float* crow = Cb + (size_t)(m0 + 8 * lh) * 2048 + n0 + ll;
#pragma unroll
  for (int r = 0; r < 8; ++r) {
    crow[(size_t)r * 2048 + 0]  = acc0[r];
    crow[(size_t)r * 2048 + 16] = acc1[r];
    crow[(size_t)r * 2048 + 32] = acc2[r];
    crow[(size_t)r * 2048 + 48] = acc3[r];
  }
}

// ======================= Stats pipeline kernels =======================

// xm[s*64+d] = mean over 32 batch slices; trans: element at x[b*131072 + d*2048 + s]
__global__ void mean_kernel(const float* __restrict__ x, float* __restrict__ xm, int trans) {
  unsigned idx = blockIdx.x * blockDim.x + threadIdx.x;
  if (idx >= MAT_ELEMS) return;
  unsigned s = idx >> 6, d = idx & 63u;
  size_t off = trans ? ((size_t)d * 2048 + s) : (size_t)idx;
  float sum = 0.0f;
  for (int b = 0; b < 32; ++b) sum += x[(size_t)b * MAT_ELEMS + off];
  xm[idx] = sum * (1.0f / 32.0f);
}

// G = xm^T @ xm  (64x64); grid 16 x 256, one entry per thread
__global__ void gram_kernel(const float* __restrict__ xm, float* __restrict__ G) {
  unsigned idx = blockIdx.x * blockDim.x + threadIdx.x;
  if (idx >= 4096u) return;
  unsigned i = idx >> 6, j = idx & 63u;
  float acc = 0.0f;
  for (int s = 0; s < 2048; ++s) acc += xm[s * 64 + i] * xm[s * 64 + j];
  G[idx] = acc;
}

// Cyclic Jacobi eigensolver on 64x64 symmetric G; outputs P = V16 @ V16^T (top-16 projector).
__global__ __launch_bounds__(64) void jacobi_kernel(const float* __restrict__ G, float* __restrict__ P) {
  __shared__ float A[64][64];
  __shared__ float V[64][64];
  __shared__ float cs[2];
  __shared__ int ord[64];
  const int t = threadIdx.x;
  for (int j = 0; j < 64; ++j) { A[t][j] = G[t * 64 + j]; V[t][j] = (t == j) ? 1.0f : 0.0f; }
  __syncthreads();
  for (int sweep = 0; sweep < 10; ++sweep) {
    for (int p = 0; p < 63; ++p) {
      for (int q = p + 1; q < 64; ++q) {
        if (t == 0) {
          float apq = A[p][q];
          float c = 1.0f, s = 0.0f;
          if (fabsf(apq) > 1e-20f) {
            float tau = (A[q][q] - A[p][p]) / (2.0f * apq);
            float tt  = (tau >= 0.0f ? 1.0f : -1.0f) / (fabsf(tau) + sqrtf(1.0f + tau * tau));
            c = 1.0f / sqrtf(1.0f + tt * tt);
            s = tt * c;
          }
          cs[0] = c; cs[1] = s;
        }
        __syncthreads();
        float c = cs[0], s = cs[1];
        // rows p,q  (A <- J^T A)
        float apt = A[p][t], aqt = A[q][t];
        A[p][t] = c * apt - s * aqt;
        A[q][t] = s * apt + c * aqt;
        __syncthreads();
        // cols p,q  (A <- A J), V <- V J
        float atp = A[t][p], atq = A[t][q];
        A[t][p] = c * atp - s * atq;
        A[t][q] = s * atp + c * atq;
        float vtp = V[t][p], vtq = V[t][q];
        V[t][p] = c * vtp - s * vtq;
        V[t][q] = s * vtp + c * vtq;
        __syncthreads();
      }
    }
  }
  if (t == 0) {   // top-16 by eigenvalue (diagonal), selection sort
    for (int i = 0; i < 64; ++i) ord[i] = i;
    for (int i = 0; i < 16; ++i) {
      int best = i;
      for (int j = i + 1; j < 64; ++j)
        if (A[ord[j]][ord[j]] > A[ord[best]][ord[best]]) best = j;
      int tmp = ord[i]; ord[i] = ord[best]; ord[best] = tmp;
    }
  }
  __syncthreads();
  for (int j = 0; j < 64; ++j) {
    float acc = 0.0f;
    for (int k = 0; k < 16; ++k) acc += V[t][ord[k]] * V[j][ord[k]];
    P[t * 64 + j] = acc;
  }
}

// LR = xm @ P   (2048x64)
__global__ void lr_kernel(const float* __restrict__ xm, const float* __restrict__ P,
                          float* __restrict__ LR) {
  unsigned idx = blockIdx.x * blockDim.x + threadIdx.x;
  if (idx >= MAT_ELEMS) return;
  unsigned s = idx >> 6, d = idx & 63u;
  float acc = 0.0f;
  for (int j = 0; j < 64; ++j) acc += xm[s * 64 + j] * P[j * 64 + d];
  LR[idx] = acc;
}

// scalar block: [0]=lo [1]=hi(max|x|, bits-maxed) [2]=t [3]=scale [4]=count(u32) [5]=submax(bits)
__global__ void init_scalars(float* sc) {
  sc[0] = 0.0f;
  ((unsigned*)sc)[1] = 0u;
  sc[2] = 0.0f; sc[3] = 0.0f;
  ((unsigned*)sc)[4] = 0u;
  ((unsigned*)sc)[5] = 0u;
}

__global__ void maxabs_kernel(const float* __restrict__ x, float* sc) {
  __shared__ float red[256];
  float m = 0.0f;
  for (unsigned i = blockIdx.x * blockDim.x + threadIdx.x; i < TOT_ELEMS; i += gridDim.x * blockDim.x)
    m = fmaxf(m, fabsf(x[i]));
  red[threadIdx.x] = m; __syncthreads();
  for (int s = 128; s > 0; s >>= 1) {
    if ((int)threadIdx.x < s) red[threadIdx.x] = fmaxf(red[threadIdx.x], red[threadIdx.x + s]);
    __syncthreads();
  }
  if (threadIdx.x == 0) atomicMax((unsigned*)sc + 1, __float_as_uint(red[0]));
}

__global__ void count_above(const float* __restrict__ x, float* sc) {
  __shared__ unsigned red[256];
  float mid = 0.5f * (sc[0] + sc[1]);
  unsigned c = 0;
  for (unsigned i = blockIdx.x * blockDim.x + threadIdx.x; i < TOT_ELEMS; i += gridDim.x * blockDim.x)
    c += (fabsf(x[i]) > mid) ? 1u : 0u;
  red[threadIdx.x] = c; __syncthreads();
  for (int s = 128; s > 0; s >>= 1) {
    if ((int)threadIdx.x < s) red[threadIdx.x] += red[threadIdx.x + s];
    __syncthreads();
  }
  if (threadIdx.x == 0) atomicAdd((unsigned*)sc + 4, red[0]);
}

__global__ void bisect_update(float* sc, unsigned target) {
  float lo = sc[0], hi = sc[1];
  float mid = 0.5f * (lo + hi);
  unsigned c = ((unsigned*)sc)[4];
  if (c > target) lo = mid; else hi = mid;
  sc[0] = lo; sc[1] = hi; sc[2] = 0.5f * (lo + hi);
  ((unsigned*)sc)[4] = 0u;
}

// max |where(|x|>t,0,x) - LR| -> sc[5] (bits)
__global__ void submax_kernel(const float* __restrict__ x, const float* __restrict__ LR,
                              float* sc, int trans) {
  __shared__ float red[256];
  unsigned idx = blockIdx.x * blockDim.x + threadIdx.x;   // grid covers TOT exactly
  float t = sc[2];
  unsigned b = idx >> 17, r = idx & (MAT_ELEMS - 1u);
  unsigned s = r >> 6, d = r & 63u;
  size_t off = trans ? ((size_t)b * MAT_ELEMS + (size_t)d * 2048 + s) : (size_t)idx;
  float v = x[off];
  float a = (fabsf(v) > t) ? 0.0f : v;
  float sub = fabsf(a - LR[r]);
  red[threadIdx.x] = sub; __syncthreads();
  for (int w = 128; w > 0; w >>= 1) {
    if ((int)threadIdx.x < w) red[threadIdx.x] = fmaxf(red[threadIdx.x], red[threadIdx.x + w]);
    __syncthreads();
  }
  if (threadIdx.x == 0) atomicMax((unsigned*)sc + 5, __float_as_uint(red[0]));
}

__global__ void finalize_scale(float* sc) {
  sc[3] = __uint_as_float(((unsigned*)sc)[5]) / QMAXF;
}

// x_hat = outlier + clip(rint(sub/scale))*scale + LR ; written in source layout
__global__ void decompress_kernel(const float* __restrict__ x, const float* __restrict__ LR,
                                  const float* sc, float* __restrict__ out, int trans) {
  unsigned idx = blockIdx.x * blockDim.x + threadIdx.x;
  float t = sc[2], scale = sc[3];
  unsigned b = idx >> 17, r = idx & (MAT_ELEMS - 1u);
  unsigned s = r >> 6, d = r & 63u;
  size_t off = trans ? ((size_t)b * MAT_ELEMS + (size_t)d * 2048 + s) : (size_t)idx;
  float v  = x[off];
  float lr = LR[r];
  float outl = (fabsf(v) > t) ? v : 0.0f;
  float sub  = v - outl - lr;
  float q = rintf(sub / scale);
  q = fminf(QMAXF, fmaxf(-QMAXF, q));
  out[off] = outl + q * scale + lr;
}

// ======================= Host-side launch =======================
extern "C" void kernel_launch(void* const* d_in, const int* in_sizes, int n_in,
                              void* d_out, int out_size, void* d_ws, size_t ws_size,
                              hipStream_t stream) {
  (void)in_sizes; (void)n_in; (void)out_size; (void)ws_size;
  const float* x1 = (const float*)d_in[0];   // [2,16,2048,64]
  const float* x2 = (const float*)d_in[1];   // [2,16,64,2048]
  float* out   = (float*)d_out;
  float* res   = out;                         // [2,16,2048,2048]
  float* x1hat = out + GEMM_ELEMS;            // [2,16,2048,64]
  float* x2hat = x1hat + (size_t)TOT_ELEMS;   // [2,16,64,2048]

  float* ws  = (float*)d_ws;
  float* xm1 = ws;                 // 131072
  float* xm2 = ws + 131072;        // 131072
  float* LR1 = ws + 262144;        // 131072
  float* LR2 = ws + 393216;        // 131072
  float* G   = ws + 524288;        // 4096
  float* P1  = ws + 528384;        // 4096
  float* P2  = ws + 532480;        // 4096
  float* sc1 = ws + 536576;        // 8
  float* sc2 = ws + 536584;        // 8

  // --- main product via WMMA f32 ---
  gemm_f32_wmma<<<dim3(32, 16, 32), 256, 0, stream>>>(x1, x2, res);

  struct Job { const float* x; int trans; float* xm; float* P; float* LR; float* sc; float* oh; };
  Job jobs[2] = {
    { x1, 0, xm1, P1, LR1, sc1, x1hat },
    { x2, 1, xm2, P2, LR2, sc2, x2hat },
  };

  for (int j = 0; j < 2; ++j) {
    Job& jb = jobs[j];
    mean_kernel<<<512, 256, 0, stream>>>(jb.x, jb.xm, jb.trans);
    gram_kernel<<<16, 256, 0, stream>>>(jb.xm, G);
    jacobi_kernel<<<1, 64, 0, stream>>>(G, jb.P);
    lr_kernel<<<512, 256, 0, stream>>>(jb.xm, jb.P, jb.LR);
    init_scalars<<<1, 1, 0, stream>>>(jb.sc);
    maxabs_kernel<<<1024, 256, 0, stream>>>(jb.x, jb.sc);
    for (int it = 0; it < 30; ++it) {
      count_above<<<1024, 256, 0, stream>>>(jb.x, jb.sc);
      bisect_update<<<1, 1, 0, stream>>>(jb.sc, QTARGET);
    }
    submax_kernel<<<16384, 256, 0, stream>>>(jb.x, jb.LR, jb.sc, jb.trans);
    finalize_scale<<<1, 1, 0, stream>>>(jb.sc);
    decompress_kernel<<<16384, 256, 0, stream>>>(jb.x, jb.LR, jb.sc, jb.oh, jb.trans);
  }
}